// TaskHeads_89206470738344
// MI455X (gfx1250) — compile-verified
//
#include <hip/hip_runtime.h>

#define BATCH 8
#define CHN   32
#define VOX   (64 * 64 * 64)
#define NC0   3
#define NC1   2
#define OUT0_ELEMS (BATCH * NC0 * VOX)
#define OUT1_ELEMS (BATCH * NC1 * VOX)

typedef __attribute__((ext_vector_type(2))) float v2f;
typedef __attribute__((ext_vector_type(8))) float v8f;

// Main streaming kernel: one pass over x, both heads via f32 WMMA,
// masked pred stores + CE partial sums.
__global__ __launch_bounds__(256) void taskheads_main(
    const float* __restrict__ x, const int* __restrict__ task_idx,
    const int* __restrict__ y, const float* __restrict__ W0,
    const float* __restrict__ b0, const float* __restrict__ W1,
    const float* __restrict__ b1, float* __restrict__ out0,
    float* __restrict__ out1, float* __restrict__ ce_acc)
{
  const int lane = threadIdx.x & 31;   // wave32
  const int wave = threadIdx.x >> 5;   // 8 waves / block
  const int half = lane >> 4;          // K-half select (A/B layout)
  const int l16  = lane & 15;          // M (class) for A, N (voxel) for B/D
  const int b    = blockIdx.y;

  // ---- Preload A tiles: 16x4 f32 per K-slice, class m = l16 (rows >=5 zero)
  v2f a[8];
  {
    const int m = l16;
    #pragma unroll
    for (int s = 0; s < 8; ++s) {
      const int k0 = s * 4 + half * 2;
      float w0v = 0.f, w1v = 0.f;
      if (m < NC0) {
        w0v = W0[m * CHN + k0];
        w1v = W0[m * CHN + k0 + 1];
      } else if (m < NC0 + NC1) {
        w0v = W1[(m - NC0) * CHN + k0];
        w1v = W1[(m - NC0) * CHN + k0 + 1];
      }
      a[s].x = w0v;
      a[s].y = w1v;
    }
  }
  const float bias0 = b0[0], bias1 = b0[1], bias2 = b0[2];
  const float bias3 = b1[0], bias4 = b1[1];
  const int   task  = task_idx[b];
  const float m0f   = (task == 0) ? 1.f : 0.f;
  const float m1f   = (task == 1) ? 1.f : 0.f;

  const float* xb = x + (size_t)b * CHN * VOX;
  float acc0 = 0.f, acc1 = 0.f;

  // Each wave: 8 groups of 16 consecutive voxels (128 voxels / wave / block-pass)
  const int base = blockIdx.x * 1024 + wave * 128;
  #pragma unroll 1
  for (int g = 0; g < 8; ++g) {
    const int vb = base + g * 16;
    v8f c = {};
    #pragma unroll
    for (int s = 0; s < 8; ++s) {
      const int k = s * 4 + half * 2;
      v2f bv;
      bv.x = xb[(size_t)k       * VOX + vb + l16];
      bv.y = xb[(size_t)(k + 1) * VOX + vb + l16];
      // D = A(16x4) x B(4x16) + C ; chained over 8 K-slices (K=32 total)
      c = __builtin_amdgcn_wmma_f32_16x16x4_f32(false, a[s], false, bv,
                                                (short)0, c, false, false);
    }

    // Lanes 0-15 hold all logits for voxel (vb + l16) in c[0..4]
    if (half == 0) {
      const int v = vb + l16;
      const float z0 = c[0] + bias0, z1 = c[1] + bias1, z2 = c[2] + bias2;
      const float z3 = c[3] + bias3, z4 = c[4] + bias4;

      // routed (masked) predictions — coalesced 64B row stores
      out0[((size_t)b * NC0 + 0) * VOX + v] = z0 * m0f;
      out0[((size_t)b * NC0 + 1) * VOX + v] = z1 * m0f;
      out0[((size_t)b * NC0 + 2) * VOX + v] = z2 * m0f;
      out1[((size_t)b * NC1 + 0) * VOX + v] = z3 * m1f;
      out1[((size_t)b * NC1 + 1) * VOX + v] = z4 * m1f;

      // per-voxel softmax CE for both heads (y in {0,1})
      const int yv = y[(size_t)b * VOX + v];
      const float mx0  = fmaxf(fmaxf(z0, z1), z2);
      const float lse0 = mx0 + __logf(__expf(z0 - mx0) + __expf(z1 - mx0) +
                                      __expf(z2 - mx0));
      acc0 += lse0 - ((yv == 0) ? z0 : z1);
      const float mx1  = fmaxf(z3, z4);
      const float lse1 = mx1 + __logf(__expf(z3 - mx1) + __expf(z4 - mx1));
      acc1 += lse1 - ((yv == 0) ? z3 : z4);
    }
  }

  // wave reduction (lanes 16-31 contribute 0), then one atomic per wave
  #pragma unroll
  for (int off = 16; off >= 1; off >>= 1) {
    acc0 += __shfl_xor(acc0, off, 32);
    acc1 += __shfl_xor(acc1, off, 32);
  }
  if (lane == 0) {
    atomicAdd(&ce_acc[b * 2 + 0], acc0);
    atomicAdd(&ce_acc[b * 2 + 1], acc1);
  }
}

__global__ void init_ws(float* ce_acc) {
  if (threadIdx.x < BATCH * 2) ce_acc[threadIdx.x] = 0.f;
}

__global__ void finalize(const int* __restrict__ task_idx,
                         const float* __restrict__ ce_acc,
                         float* __restrict__ total_out) {
  if (threadIdx.x != 0) return;
  float n0 = 0.f, n1 = 0.f;
  for (int b = 0; b < BATCH; ++b) {
    if (task_idx[b] == 0) n0 += 1.f;
    if (task_idx[b] == 1) n1 += 1.f;
  }
  float s0 = 0.f, s1 = 0.f;
  for (int b = 0; b < BATCH; ++b) {
    const float per0 = ce_acc[b * 2 + 0] / (float)VOX;  // mean over voxels
    const float per1 = ce_acc[b * 2 + 1] / (float)VOX;
    if (task_idx[b] == 0) s0 += per0;
    if (task_idx[b] == 1) s1 += per1;
  }
  const float loss0 = s0 / fmaxf(n0, 1.f);
  const float loss1 = s1 / fmaxf(n1, 1.f);
  const float w0 = 3.f * ((n0 > 0.f) ? 1.f : 0.f);
  const float w1 = 2.f * ((n1 > 0.f) ? 1.f : 0.f);
  *total_out = (loss0 * w0 + loss1 * w1) / fmaxf(w0 + w1, 1.f);
}

extern "C" void kernel_launch(void* const* d_in, const int* in_sizes, int n_in,
                              void* d_out, int out_size, void* d_ws, size_t ws_size,
                              hipStream_t stream) {
  const float* x        = (const float*)d_in[0];
  const int*   task_idx = (const int*)d_in[1];
  const int*   y        = (const int*)d_in[2];
  const float* W0       = (const float*)d_in[3];
  const float* b0       = (const float*)d_in[4];
  const float* W1       = (const float*)d_in[5];
  const float* b1       = (const float*)d_in[6];

  float* out0  = (float*)d_out;
  float* out1  = out0 + OUT0_ELEMS;
  float* total = out1 + OUT1_ELEMS;
  float* ce_acc = (float*)d_ws;  // 16 floats

  init_ws<<<1, 32, 0, stream>>>(ce_acc);

  dim3 grid(VOX / 1024, BATCH);   // 8 waves/block * 8 groups * 16 voxels = 1024
  taskheads_main<<<grid, 256, 0, stream>>>(x, task_idx, y, W0, b0, W1, b1,
                                           out0, out1, ce_acc);

  finalize<<<1, 32, 0, stream>>>(task_idx, ce_acc, total);
}